// ModuleFilterInterpolation_36799279792300
// MI455X (gfx1250) — compile-verified
//
#include <hip/hip_runtime.h>
#include <hip/hip_bf16.h>

// ModuleFilterInterpolation: B=4, C=3, H=512, W=960, K=4 (K2=16).
// HBM-bound (~190 MB/call -> ~8 us at 23.3 TB/s). One thread per pixel,
// 5x5 effective-weight reformulation, async-to-LDS staging of the 16
// streaming filter planes (gfx1250 ASYNCcnt path), NT hints on single-use
// streams so teninput stays hot in the 192 MB L2 for the 25x gather reuse.

#define BB 4
#define CC 3
#define HH 512
#define WW 960

#if defined(__AMDGCN__) && __has_builtin(__builtin_amdgcn_global_load_async_to_lds_b32)
#define USE_ASYNC_LDS 1
#else
#define USE_ASYNC_LDS 0
#endif

__global__ __launch_bounds__(256) void filt_interp_kernel(
    const float* __restrict__ input,   // [B,C,H,W]
    const float* __restrict__ flow,    // [B,2,H,W]
    const float* __restrict__ filt,    // [B,16,H,W]
    float* __restrict__ out)           // [B,C,H,W]
{
    constexpr int Wd = WW, Hd = HH, HW = Hd * Wd, CHW = CC * HW;
    constexpr int NPIX = BB * HW;

    const int tid = threadIdx.x;
    const int p = blockIdx.x * 256 + tid;
    if (p >= NPIX) return;

    const int b = p / HW;
    const int r = p - b * HW;
    const int y = r / Wd;
    const int x = r - y * Wd;

    const float* fptr = filt + (size_t)b * 16 * HW + r;

    float F[4][4];
#if USE_ASYNC_LDS
    // Stage the 16 coalesced filter-plane loads into LDS with the gfx1250
    // async path: no dest VGPRs held across the latency window, landed in
    // 16 KB of LDS, read back bank-conflict-free ([k][lane] layout).
    __shared__ float sF[16 * 256];
    {
        typedef __attribute__((address_space(1))) int gint_t;
        typedef __attribute__((address_space(3))) int lint_t;
        #pragma unroll
        for (int k = 0; k < 16; ++k) {
            __builtin_amdgcn_global_load_async_to_lds_b32(
                (gint_t*)(fptr + k * HW),
                (lint_t*)(&sF[k * 256 + tid]),
                0, 0);
        }
    }
    // Overlap: compute flow/validity/weights while async loads are in flight.
#endif

    const float* flB = flow + (size_t)b * 2 * HW;
    const float fx = __builtin_nontemporal_load(flB + r);
    const float fy = __builtin_nontemporal_load(flB + HW + r);

    const float x2 = (float)x + fx;
    const float y2 = (float)y + fy;
    const bool valid = (x2 >= 0.f) && (x2 <= (float)(Wd - 1)) &&
                       (y2 >= 0.f) && (y2 <= (float)(Hd - 1)) &&
                       (fabsf(fx) < 0.5f * (float)Wd) &&
                       (fabsf(fy) < 0.5f * (float)Hd);
    const float vmask = valid ? 1.f : 0.f;

    // Clamp before floor: invalid lanes stay in-range (no OOB gathers, no
    // Inf*0=NaN); for valid lanes this is exactly the reference.
    const float xc = fminf(fmaxf(x2, 0.f), (float)(Wd - 1));
    const float yc = fminf(fmaxf(y2, 0.f), (float)(Hd - 1));
    const int ix = (int)floorf(xc);
    const int iy = (int)floorf(yc);
    const float alpha = xc - (float)ix;
    const float beta  = yc - (float)iy;

    const float w00 = (1.f - alpha) * (1.f - beta);  // TL
    const float w01 = alpha * (1.f - beta);          // TR
    const float w10 = (1.f - alpha) * beta;          // BL
    const float w11 = alpha * beta;                  // BR

    int xs[5], ys[5];
    #pragma unroll
    for (int i = 0; i < 5; ++i) {
        xs[i] = min(max(ix - 1 + i, 0), Wd - 1);
        ys[i] = min(max(iy - 1 + i, 0), Hd - 1);
    }

#if USE_ASYNC_LDS
#if __has_builtin(__builtin_amdgcn_s_wait_asynccnt)
    __builtin_amdgcn_s_wait_asynccnt(0);
#else
    asm volatile("s_wait_asynccnt 0x0" ::: "memory");
#endif
    asm volatile("" ::: "memory");  // keep LDS reads below the wait
    #pragma unroll
    for (int k = 0; k < 16; ++k) F[k >> 2][k & 3] = sF[k * 256 + tid];
#else
    #pragma unroll
    for (int k = 0; k < 16; ++k)
        F[k >> 2][k & 3] = __builtin_nontemporal_load(fptr + k * HW);
#endif

    // Effective 5x5 weights: 4x4 filter "convolved" with the 2x2 bilinear
    // kernel (exact under monotone clipping: clip(i0+d) enumerates the same
    // 5 rows/cols the reference taps touch).
    float Wt[5][5];
    #pragma unroll
    for (int j = 0; j < 5; ++j) {
        #pragma unroll
        for (int i = 0; i < 5; ++i) {
            float s = 0.f;
            if (j < 4 && i < 4) s = fmaf(w00, F[j][i], s);
            if (j < 4 && i > 0) s = fmaf(w01, F[j][i - 1], s);
            if (j > 0 && i < 4) s = fmaf(w10, F[j - 1][i], s);
            if (j > 0 && i > 0) s = fmaf(w11, F[j - 1][i - 1], s);
            Wt[j][i] = s;
        }
    }

    // 25 gathers per channel; weights/addresses shared across the 3 channels.
    const float* in0 = input + (size_t)b * CHW;
    const float* in1 = in0 + HW;
    const float* in2 = in1 + HW;
    float a0 = 0.f, a1 = 0.f, a2 = 0.f;
    #pragma unroll
    for (int j = 0; j < 5; ++j) {
        const int rb = ys[j] * Wd;
        #pragma unroll
        for (int i = 0; i < 5; ++i) {
            const int idx = rb + xs[i];
            const float wv = Wt[j][i];
            a0 = fmaf(wv, in0[idx], a0);
            a1 = fmaf(wv, in1[idx], a1);
            a2 = fmaf(wv, in2[idx], a2);
        }
    }

    float* ob = out + (size_t)b * CHW + r;
    __builtin_nontemporal_store(a0 * vmask, ob);
    __builtin_nontemporal_store(a1 * vmask, ob + HW);
    __builtin_nontemporal_store(a2 * vmask, ob + 2 * HW);
}

extern "C" void kernel_launch(void* const* d_in, const int* in_sizes, int n_in,
                              void* d_out, int out_size, void* d_ws, size_t ws_size,
                              hipStream_t stream) {
    (void)in_sizes; (void)n_in; (void)d_ws; (void)ws_size; (void)out_size;
    const float* teninput  = (const float*)d_in[0];
    const float* tenflow   = (const float*)d_in[1];
    const float* tenfilter = (const float*)d_in[2];
    float* out = (float*)d_out;

    const int npix = BB * HH * WW;                 // 1,966,080
    const int block = 256;                         // 8 wave32 per block
    const int grid = (npix + block - 1) / block;   // 7680
    filt_interp_kernel<<<grid, block, 0, stream>>>(teninput, tenflow, tenfilter, out);
}